// VelocityExtractor_38414187495446
// MI455X (gfx1250) — compile-verified
//
#include <hip/hip_runtime.h>
#include <math.h>

#define OUT_SZ   224
#define NBINS    8
#define IMG_W    256
#define IMG_H    256
#define STRIP    28                 // output rows per strip
#define NSTRIP   (OUT_SZ / STRIP)   // 8
#define CAP_ROWS 24                 // LDS row window per channel (48KB per buffer)
#define PI_F     3.14159265358979323846f

// ---- CDNA5 async global->LDS copy of 16 bytes (per-lane) ----------------
__device__ __forceinline__ void async_cp16(const float* gp, float* lp) {
#if defined(__AMDGCN__) && __has_builtin(__builtin_amdgcn_global_load_async_to_lds_b128)
    typedef int v4i_t __attribute__((ext_vector_type(4)));
    typedef __attribute__((address_space(1))) v4i_t GV4;   // global
    typedef __attribute__((address_space(3))) v4i_t LV4;   // LDS (32-bit offset)
    __builtin_amdgcn_global_load_async_to_lds_b128(
        (GV4*)(unsigned long long)gp,
        (LV4*)(unsigned int)(unsigned long long)lp,
        0, 0);
#else
    float4 t = *(const float4*)gp;
    *(float4*)lp = t;
#endif
}

__device__ __forceinline__ void wait_async0() {
#if defined(__AMDGCN__)
#if __has_builtin(__builtin_amdgcn_s_wait_asynccnt)
    __builtin_amdgcn_s_wait_asynccnt(0);
#else
    asm volatile("s_wait_asynccnt 0" ::: "memory");
#endif
#endif
}

// image-row window touched by strip s's (clamped) sample ys
__device__ __forceinline__ void strip_window(float y1f, float bin_h, int s,
                                             int* ry0, int* nrows) {
    const int r0 = s * STRIP;
    float ylo = y1f + ((float)r0 + 0.25f) * bin_h;
    float yhi = y1f + ((float)r0 + (float)(STRIP - 1) + 0.75f) * bin_h;
    ylo = fminf(fmaxf(ylo, 0.0f), (float)(IMG_H - 1));
    yhi = fminf(fmaxf(yhi, 0.0f), (float)(IMG_H - 1));
    *ry0 = (int)floorf(ylo);
    const int ry1 = min((int)floorf(yhi) + 1, IMG_H - 1);
    *nrows = ry1 - *ry0 + 1;
}

// issue async DMA of rows [ry0, ry0+nrows) (both channels) into sbuf
__device__ __forceinline__ void stage_async(const float* gimg, float* sbuf,
                                            int ry0, int nrows, int tid) {
    const int chunks = nrows * (IMG_W / 4);   // 16B chunks per channel
    for (int c = tid; c < 2 * chunks; c += 256) {
        const int ch  = (c >= chunks) ? 1 : 0;
        const int cc  = c - ch * chunks;
        const int row = cc >> 6;
        const int xo  = (cc & 63) << 2;
        const float* gp = gimg + ((size_t)ch * IMG_H + (size_t)(ry0 + row)) * IMG_W + xo;
        float*       lp = &sbuf[(ch * CAP_ROWS + row) * IMG_W + xo];
        async_cp16(gp, lp);
    }
}

// ---- branchless ROI-align sampling + histogram over one 28-row strip ----
// __forceinline__ keeps pointer provenance visible: LDS caller -> ds_load,
// global caller -> global_load (no flat ops).
__device__ __forceinline__ void sample_strip(
    const float* c0, const float* c1, int rowoff,
    int r0, int tid, float x1f, float y1f, float bin_w, float bin_h,
    float ws[NBINS], float cn[NBINS])
{
    for (int i = tid; i < STRIP * OUT_SZ; i += 256) {
        const int lrow = i / OUT_SZ;
        const int px   = i - lrow * OUT_SZ;
        const int py   = r0 + lrow;

        int yr0[2], yr1[2], xi0[2], xi1[2];
        float ly[2], hy[2], lx[2], hx[2];
#pragma unroll
        for (int j = 0; j < 2; ++j) {
            float t = y1f + ((float)py + (j ? 0.75f : 0.25f)) * bin_h;
            const bool v = (t > -1.0f) && (t < (float)IMG_H);
            t = fminf(fmaxf(t, 0.0f), (float)(IMG_H - 1));
            const float f = floorf(t);
            const int i0 = (int)f;
            ly[j] = v ? (t - f) : 0.0f;                // validity folded into
            hy[j] = v ? (1.0f - (t - f)) : 0.0f;       // the 1-D weights
            yr0[j] = (i0 - rowoff) * IMG_W;
            yr1[j] = (min(i0 + 1, IMG_H - 1) - rowoff) * IMG_W;
        }
#pragma unroll
        for (int j = 0; j < 2; ++j) {
            float t = x1f + ((float)px + (j ? 0.75f : 0.25f)) * bin_w;
            const bool v = (t > -1.0f) && (t < (float)IMG_W);
            t = fminf(fmaxf(t, 0.0f), (float)(IMG_W - 1));
            const float f = floorf(t);
            xi0[j] = (int)f;
            lx[j] = v ? (t - f) : 0.0f;
            hx[j] = v ? (1.0f - (t - f)) : 0.0f;
            xi1[j] = min(xi0[j] + 1, IMG_W - 1);
        }

        float au = 0.0f, av = 0.0f;
#pragma unroll
        for (int jy = 0; jy < 2; ++jy) {
#pragma unroll
            for (int jx = 0; jx < 2; ++jx) {
                const int i00 = yr0[jy] + xi0[jx], i01 = yr0[jy] + xi1[jx];
                const int i10 = yr1[jy] + xi0[jx], i11 = yr1[jy] + xi1[jx];
                const float w00 = hy[jy] * hx[jx], w01 = hy[jy] * lx[jx];
                const float w10 = ly[jy] * hx[jx], w11 = ly[jy] * lx[jx];
                au += w00 * c0[i00] + w01 * c0[i01] + w10 * c0[i10] + w11 * c0[i11];
                av += w00 * c1[i00] + w01 * c1[i01] + w10 * c1[i10] + w11 * c1[i11];
            }
        }
        const float u   = au * 0.25f;
        const float v   = av * 0.25f;
        const float mag = sqrtf(u * u + v * v);
        const float th  = atan2f(u, v);   // atan2(c0, c1) as in reference
        int bin = (int)floorf((th + PI_F) * ((float)NBINS / (2.0f * PI_F)));
        bin = max(0, min(bin, NBINS - 1));
#pragma unroll
        for (int b = 0; b < NBINS; ++b) {
            const bool m = (bin == b);
            ws[b] += m ? mag : 0.0f;
            cn[b] += m ? 1.0f : 0.0f;
        }
    }
}

__global__ __launch_bounds__(256) void velhist_kernel(
    const float* __restrict__ flows, const float* __restrict__ boxes,
    float* __restrict__ out, int K)
{
    __shared__ float s_img[2][2 * CAP_ROWS * IMG_W];   // double buffer, 96KB
    __shared__ float s_hist[2 * NBINS];                // [0..7]=wsum, [8..15]=count

    const int k   = blockIdx.x;
    const int tid = threadIdx.x;
    if (k >= K) return;

    if (tid < 2 * NBINS) s_hist[tid] = 0.0f;

    const float x1f = boxes[k * 5 + 1];
    const float y1f = boxes[k * 5 + 2];
    const float x2f = boxes[k * 5 + 3];
    const float y2f = boxes[k * 5 + 4];
    const int   bi  = (int)boxes[k * 5 + 0];
    const float bin_w = fmaxf(x2f - x1f, 1.0f) / (float)OUT_SZ;
    const float bin_h = fmaxf(y2f - y1f, 1.0f) / (float)OUT_SZ;

    const float* __restrict__ gimg = flows + (size_t)bi * 2 * IMG_H * IMG_W;

    float ws[NBINS], cn[NBINS];
#pragma unroll
    for (int b = 0; b < NBINS; ++b) { ws[b] = 0.0f; cn[b] = 0.0f; }

    // prologue: stage strip 0 into buffer 0
    int  cur_ry0, cur_n;
    strip_window(y1f, bin_h, 0, &cur_ry0, &cur_n);
    bool cur_lds = (cur_n <= CAP_ROWS);
    if (cur_lds) {
        stage_async(gimg, s_img[0], cur_ry0, cur_n, tid);
        wait_async0();
    }
    __syncthreads();   // buffer 0 ready everywhere; also orders s_hist zeroing

    for (int s = 0; s < NSTRIP; ++s) {
        // software pipeline: kick off strip s+1's DMA before sampling strip s
        int  nxt_ry0 = 0, nxt_n = 0;
        bool nxt_lds = false;
        if (s + 1 < NSTRIP) {
            strip_window(y1f, bin_h, s + 1, &nxt_ry0, &nxt_n);
            nxt_lds = (nxt_n <= CAP_ROWS);
            if (nxt_lds) {
                // alternate buffer last read in iteration s-1; that iteration's
                // trailing barrier makes this overwrite safe
                stage_async(gimg, s_img[(s + 1) & 1], nxt_ry0, nxt_n, tid);
            } else {
                __builtin_prefetch(gimg + (size_t)nxt_ry0 * IMG_W, 0, 1);
            }
        }

        const int r0 = s * STRIP;
        if (cur_lds) {
            const float* buf = s_img[s & 1];
            sample_strip(buf, buf + CAP_ROWS * IMG_W, cur_ry0,
                         r0, tid, x1f, y1f, bin_w, bin_h, ws, cn);
        } else {
            sample_strip(gimg, gimg + IMG_H * IMG_W, 0,
                         r0, tid, x1f, y1f, bin_w, bin_h, ws, cn);
        }

        wait_async0();     // this wave's DMA for strip s+1 has landed
        __syncthreads();   // all waves' DMA landed; all waves done reading strip s

        cur_ry0 = nxt_ry0;
        cur_n   = nxt_n;
        cur_lds = nxt_lds;
    }

    // block-level histogram merge
#pragma unroll
    for (int b = 0; b < NBINS; ++b) {
        atomicAdd(&s_hist[b],         ws[b]);
        atomicAdd(&s_hist[NBINS + b], cn[b]);
    }
    __syncthreads();
    if (tid < NBINS) {
        const float c = s_hist[NBINS + tid];
        out[k * NBINS + tid] = (c != 0.0f) ? (s_hist[tid] / c) : 0.0f;
    }
}

extern "C" void kernel_launch(void* const* d_in, const int* in_sizes, int n_in,
                              void* d_out, int out_size, void* d_ws, size_t ws_size,
                              hipStream_t stream) {
    const float* flows = (const float*)d_in[0];
    const float* boxes = (const float*)d_in[1];
    float*       out   = (float*)d_out;
    const int K = in_sizes[1] / 5;   // boxes are [K,5]
    velhist_kernel<<<dim3(K), dim3(256), 0, stream>>>(flows, boxes, out, K);
}